// IsolatedNodeExpert_58308476011148
// MI455X (gfx1250) — compile-verified
//
#include <hip/hip_runtime.h>

typedef float v2f __attribute__((ext_vector_type(2)));
typedef float v8f __attribute__((ext_vector_type(8)));

#define NB   8
#define NN   50000
#define TT   64
#define HID  256
#define HOR  24
#define EPSI 0.001f

// ---------------- zero accumulators (deg_w, deg_cnt, agg) ------------------
__global__ void k_zero(float* __restrict__ p, int total) {
    int i = blockIdx.x * blockDim.x + threadIdx.x;
    if (i < total) p[i] = 0.0f;
}

// ---------------- edge pass: weighted out-degree + in-degree count ---------
__global__ void k_edges(const int* __restrict__ ei, const float* __restrict__ ew,
                        float* __restrict__ degw, float* __restrict__ degc, int E) {
    int e = blockIdx.x * blockDim.x + threadIdx.x;
    if (e >= E) return;
    int s = ei[e];
    int d = ei[E + e];
    atomicAdd(degw + s, ew[e]);
    atomicAdd(degc + d, 1.0f);
}

// ---------------- dinv[n] = rsqrt(indeg + 1)  (self-loop included) ---------
__global__ void k_dinv(const float* __restrict__ degc, float* __restrict__ dinv) {
    int n = blockIdx.x * blockDim.x + threadIdx.x;
    if (n < NN) dinv[n] = rsqrtf(degc[n] + 1.0f);
}

// ---------------- mean over T=64 and xw = mean * iso -----------------------
// 16 lanes per node, each lane loads one float4 (512B coalesced per wave).
__global__ void k_meanxw(const float* __restrict__ x, const float* __restrict__ degw,
                         float* __restrict__ xw) {
    int gid  = blockIdx.x * blockDim.x + threadIdx.x;
    int node = gid >> 4;          // flat row in [0, B*N)
    int l16  = gid & 15;
    const float4* p = reinterpret_cast<const float4*>(x + (size_t)node * TT);
    float4 v = p[l16];
    float s = v.x + v.y + v.z + v.w;
    // xor masks 1,2,4,8 never flip bit 4 -> independent 16-lane reductions
    s += __shfl_xor(s, 1);
    s += __shfl_xor(s, 2);
    s += __shfl_xor(s, 4);
    s += __shfl_xor(s, 8);
    if (l16 == 0) {
        int n = node % NN;
        float iso = 1.0f / (degw[n] + EPSI);
        xw[node] = (s * (1.0f / (float)TT)) * iso;
    }
}

// ---------------- SpMV: agg[b, dst] += xw[b, src] * dinv[s]*dinv[d] --------
__global__ void k_spmv(const int* __restrict__ ei, const float* __restrict__ dinv,
                       const float* __restrict__ xw, float* __restrict__ agg, int E) {
    int e = blockIdx.x * blockDim.x + threadIdx.x;
    if (e >= E) return;
    int s = ei[e];
    int d = ei[E + e];
    float nrm = dinv[s] * dinv[d];
#pragma unroll
    for (int b = 0; b < NB; ++b)
        atomicAdd(agg + (size_t)b * NN + d, xw[(size_t)b * NN + s] * nrm);
}

// ---------------- collapse hidden dim: w_eff/b_eff [24] --------------------
__global__ void k_weff(const float* __restrict__ gw, const float* __restrict__ gb,
                       const float* __restrict__ pw, const float* __restrict__ pb,
                       float* __restrict__ weff, float* __restrict__ beff) {
    int t = threadIdx.x;
    if (t < HOR) {
        float a = 0.f, b = 0.f;
        for (int h = 0; h < HID; ++h) {
            float p = pw[h * HOR + t];
            a += gw[h] * p;
            b += gb[h] * p;
        }
        weff[t] = a;
        beff[t] = b + pb[t];
    }
}

// ---------------- fused rank-1 expand via V_WMMA_F32_16X16X4_F32 -----------
// out[r, k] = a[r] * w_eff[k] + b_eff[k],  a[r] = agg[r] + xw[r]*dinv[n]^2
// One wave -> 16 rows. Two WMMA tiles cover k=0..15 and k=16..23.
__global__ void k_expand(const float* __restrict__ agg, const float* __restrict__ xw,
                         const float* __restrict__ dinv, const float* __restrict__ weff,
                         const float* __restrict__ beff, float* __restrict__ out) {
    int wid  = (blockIdx.x * blockDim.x + threadIdx.x) >> 5;  // global wave id
    int lane = threadIdx.x & 31;
    int l16  = lane & 15;
    int row0 = wid * 16;                 // B*N == 400000, divisible by 16: exact grid

    // self-loop term folded in here
    int r   = row0 + l16;
    float dv = dinv[r % NN];
    float av = agg[r] + xw[r] * dv * dv;

    // A: 16x4, only K=0 column nonzero. VGPR0 lanes0-15 = K=0, lanes16-31 = K=2.
    v2f A;
    A.x = (lane < 16) ? av : 0.0f;
    A.y = 0.0f;

    // B tile0: row K=0 = w_eff[0..15]; tile1: row K=0 = w_eff[16..23] (cols 8..15 pad)
    v2f B0, B1;
    B0.x = (lane < 16) ? weff[l16] : 0.0f;
    B0.y = 0.0f;
    B1.x = (lane < 16 && l16 < 8) ? weff[16 + l16] : 0.0f;
    B1.y = 0.0f;

    // C: broadcast b_eff along rows; C/D layout -> every VGPR = b_eff[col(lane)]
    float be0 = beff[l16];
    float be1 = (l16 < 8) ? beff[16 + l16] : 0.0f;
    v8f C0, C1;
#pragma unroll
    for (int i = 0; i < 8; ++i) { C0[i] = be0; C1[i] = be1; }

    v8f D0 = __builtin_amdgcn_wmma_f32_16x16x4_f32(false, A, false, B0,
                                                   (short)0, C0, false, false);
    v8f D1 = __builtin_amdgcn_wmma_f32_16x16x4_f32(false, A, false, B1,
                                                   (short)0, C1, false, false);

    // D layout: VGPR v, lanes0-15 -> M=v, N=lane; lanes16-31 -> M=v+8, N=lane-16
    int mbase = (lane < 16) ? 0 : 8;
    float* op = out + (size_t)row0 * HOR;
#pragma unroll
    for (int v = 0; v < 8; ++v)
        op[(mbase + v) * HOR + l16] = D0[v];
    if (l16 < 8) {
#pragma unroll
        for (int v = 0; v < 8; ++v)
            op[(mbase + v) * HOR + 16 + l16] = D1[v];
    }
}

extern "C" void kernel_launch(void* const* d_in, const int* in_sizes, int n_in,
                              void* d_out, int out_size, void* d_ws, size_t ws_size,
                              hipStream_t stream) {
    const float* x    = (const float*)d_in[0];
    const int*   ei   = (const int*)  d_in[1];
    const float* ew   = (const float*)d_in[2];
    const float* gw   = (const float*)d_in[3];
    const float* gb   = (const float*)d_in[4];
    const float* pw   = (const float*)d_in[5];
    const float* pb   = (const float*)d_in[6];
    float* out = (float*)d_out;

    const int E  = in_sizes[2];           // 1,600,000
    const int BN = NB * NN;               // 400,000

    // workspace layout (floats)
    float* ws    = (float*)d_ws;
    float* degw  = ws;                    // [N]
    float* degc  = degw + NN;             // [N]
    float* dinv  = degc + NN;             // [N]
    float* xw    = dinv + NN;             // [B*N]
    float* agg   = xw + BN;               // [B*N]
    float* weff  = agg + BN;              // [32]
    float* beff  = weff + 32;             // [32]

    const int TPB = 256;

    // zero deg_w, deg_cnt (contiguous) and agg each call (graph-replay safe)
    k_zero<<<(2 * NN + TPB - 1) / TPB, TPB, 0, stream>>>(degw, 2 * NN);
    k_zero<<<(BN + TPB - 1) / TPB, TPB, 0, stream>>>(agg, BN);

    k_edges<<<(E + TPB - 1) / TPB, TPB, 0, stream>>>(ei, ew, degw, degc, E);
    k_dinv <<<(NN + TPB - 1) / TPB, TPB, 0, stream>>>(degc, dinv);

    // 16 threads per node over B*N nodes
    k_meanxw<<<(BN * 16) / TPB, TPB, 0, stream>>>(x, degw, xw);

    k_spmv<<<(E + TPB - 1) / TPB, TPB, 0, stream>>>(ei, dinv, xw, agg, E);

    k_weff<<<1, 64, 0, stream>>>(gw, gb, pw, pb, weff, beff);

    // one wave per 16 rows: BN/16 = 25000 waves -> 800000 threads -> 3125 blocks
    k_expand<<<(BN / 16) * 32 / TPB, TPB, 0, stream>>>(agg, xw, dinv, weff, beff, out);
}